// PairformerBlock_41162966565154
// MI455X (gfx1250) — compile-verified
//
#include <hip/hip_runtime.h>
#include <math.h>

typedef __attribute__((ext_vector_type(2))) float v2f;
typedef __attribute__((ext_vector_type(4))) float v4f;
typedef __attribute__((ext_vector_type(8))) float v8f;

#define NEGV (-1000000000.0f)

__device__ __forceinline__ v8f wmma4(v2f a, v2f b, v8f c) {
  // D(16x16,f32) = A(16x4,f32) * B(4x16,f32) + C   -- full fp32 matrix core
  return __builtin_amdgcn_wmma_f32_16x16x4_f32(false, a, false, b, (short)0, c,
                                               false, false);
}

__device__ __forceinline__ v8f zero8() {
  v8f z = {0.f, 0.f, 0.f, 0.f, 0.f, 0.f, 0.f, 0.f};
  return z;
}

// ---------------------------------------------------------------------------
// fp32-WMMA GEMM: C[M,N] = A[M,K] @ B[K,N] (+bias[n]) (opt ReLU).
// One wave computes a 16x64 strip (4 n-tiles): A fragment loaded once per
// k-step, reused by 4 WMMAs. Software-pipelined (double-buffered fragments)
// so next step's loads overlap current step's WMMAs. Ragged N handled
// branchlessly (clamped column * {0,1} weight). M % 16 == 0, K % 4 == 0.
// ---------------------------------------------------------------------------
__global__ __launch_bounds__(32) void gemm_f32_kernel(
    const float* __restrict__ A, int lda, const float* __restrict__ B, int ldb,
    float* __restrict__ C, int ldc, int M, int N, int K,
    const float* __restrict__ bias, int relu) {
  const int g0 = blockIdx.x << 6;  // first column of the 4-tile strip
  const int m0 = blockIdx.y << 4;
  const int lane = threadIdx.x;
  const int hi = lane >> 4;  // selects K pair {0,1} vs {2,3}
  const int r = lane & 15;

  v8f acc[4];
  int ncl[4];
  float wn[4];
#pragma unroll
  for (int t = 0; t < 4; ++t) {
    acc[t] = zero8();
    const int n = g0 + (t << 4) + r;
    const bool nv = (n < N);
    ncl[t] = nv ? n : 0;     // clamped, always-legal column
    wn[t] = nv ? 1.f : 0.f;  // branchless zeroing weight
  }

  const float* arow = A + (size_t)(m0 + r) * lda + 2 * hi;
  const float* brow = B + (size_t)(2 * hi) * ldb;
  const int ksteps = K >> 2;

  // Prologue: fragments for k-step 0.
  v2f a_c;
  a_c.x = arow[0];
  a_c.y = arow[1];
  float bx_c[4], by_c[4];
#pragma unroll
  for (int t = 0; t < 4; ++t) {
    bx_c[t] = brow[ncl[t]];
    by_c[t] = brow[ldb + ncl[t]];
  }

  for (int ks = 0; ks < ksteps; ++ks) {
    // Prefetch next k-step (clamped on the last iteration: re-reads, unused).
    const int adv = (ks + 1 < ksteps) ? 4 : 0;
    const float* arow_n = arow + adv;
    const float* brow_n = brow + (size_t)adv * ldb;
    v2f a_n;
    a_n.x = arow_n[0];
    a_n.y = arow_n[1];
    float bx_n[4], by_n[4];
#pragma unroll
    for (int t = 0; t < 4; ++t) {
      bx_n[t] = brow_n[ncl[t]];
      by_n[t] = brow_n[ldb + ncl[t]];
    }
    // Consume current fragments.
#pragma unroll
    for (int t = 0; t < 4; ++t) {
      v2f b;
      b.x = bx_c[t] * wn[t];
      b.y = by_c[t] * wn[t];
      acc[t] = wmma4(a_c, b, acc[t]);
    }
    a_c = a_n;
#pragma unroll
    for (int t = 0; t < 4; ++t) {
      bx_c[t] = bx_n[t];
      by_c[t] = by_n[t];
    }
    arow = arow_n;
    brow = brow_n;
  }

#pragma unroll
  for (int t = 0; t < 4; ++t) {
    const int n = g0 + (t << 4) + r;
    if (n < N) {
      const float bv = bias ? bias[n] : 0.f;
#pragma unroll
      for (int v = 0; v < 8; ++v) {
        const int m = m0 + v + 8 * hi;  // C layout: VGPR v holds rows v, v+8
        float val = acc[t][v] + bv;
        if (relu) val = fmaxf(val, 0.f);
        C[(size_t)m * ldc + n] = val;
      }
    }
  }
}

// ---------------------------------------------------------------------------
// Row LayerNorm: y[row] = (x-mean)/sqrt(var+1e-5)*g + b. One wave per row.
// ---------------------------------------------------------------------------
__global__ __launch_bounds__(32) void ln_kernel(const float* __restrict__ x,
                                                float* __restrict__ y,
                                                const float* __restrict__ g,
                                                const float* __restrict__ b,
                                                int D) {
  const int row = blockIdx.x;
  const int lane = threadIdx.x;
  const float* xr = x + (size_t)row * D;
  float s = 0.f;
  for (int d = lane; d < D; d += 32) s += xr[d];
#pragma unroll
  for (int o = 16; o > 0; o >>= 1) s += __shfl_xor(s, o, 32);
  const float mean = s / (float)D;
  float vs = 0.f;
  for (int d = lane; d < D; d += 32) {
    const float t = xr[d] - mean;
    vs += t * t;
  }
#pragma unroll
  for (int o = 16; o > 0; o >>= 1) vs += __shfl_xor(vs, o, 32);
  const float inv = 1.f / sqrtf(vs / (float)D + 1e-5f);
  float* yr = y + (size_t)row * D;
  for (int d = lane; d < D; d += 32) yr[d] = (xr[d] - mean) * inv * g[d] + b[d];
}

// ---------------------------------------------------------------------------
// Pair attention (flash, fp32 WMMA). One wave per (head, 16-query tile).
// qkv: [4096, 432]. S^T tiles (keys x queries) -> softmax row-reduce is one
// shfl_xor(16); P^T round-trips through 1KB LDS for the O^T accumulation.
// Pipelined: next tile's K frags + this tile's V frags are loaded during the
// softmax VALU work. All hd-padding guards are branchless.
// ---------------------------------------------------------------------------
__global__ __launch_bounds__(32) void pair_attn_kernel(
    const float* __restrict__ qkv, const float* __restrict__ pm,
    float* __restrict__ outp) {
  __shared__ float pT[256];
  const int h = blockIdx.y;
  const int i0 = blockIdx.x << 4;
  const int lane = threadIdx.x;
  const int hi = lane >> 4;
  const int r = lane & 15;
  const int ld = 432;
  const float* Q = qkv + h * 18;
  const float* V = qkv + 288 + h * 18;
  const float scale = 1.0f / sqrtf(18.0f);

  // Branchless hd=18 -> 20 padding: clamped dim offsets + {0,1} weights.
  int ox[5], oy[5];
  float wx[5], wy[5];
#pragma unroll
  for (int kk = 0; kk < 5; ++kk) {
    const int d0 = kk * 4 + 2 * hi;
    ox[kk] = (d0 < 18) ? d0 : 17;
    wx[kk] = (d0 < 18) ? 1.f : 0.f;
    oy[kk] = (d0 + 1 < 18) ? d0 + 1 : 17;
    wy[kk] = (d0 + 1 < 18) ? 1.f : 0.f;
  }
  const int dv1 = (16 + r < 18) ? 16 + r : 17;  // V d-tile-1 dim (clamped)
  const float wv1 = (16 + r < 18) ? 1.f : 0.f;

  // Preload Q^T B-fragments (queries = cols).
  v2f qf[5];
  {
    const float* qrow = Q + (size_t)(i0 + r) * ld;
#pragma unroll
    for (int kk = 0; kk < 5; ++kk) {
      qf[kk].x = qrow[ox[kk]] * wx[kk];
      qf[kk].y = qrow[oy[kk]] * wy[kk];
    }
  }

  float m_s = -1e30f, l_s = 0.f;
  v8f o0 = zero8();  // O^T rows d=0..15 (cols = queries)
  v8f o1 = zero8();  // O^T rows d=16..31 (only 16,17 stored)

  // Running pointers; prologue K fragments for tile j=0.
  const float* kptr = qkv + 144 + h * 18 + (size_t)r * ld;
  const float* vb = V;
  v2f kf[5];
#pragma unroll
  for (int kk = 0; kk < 5; ++kk) {
    kf[kk].x = kptr[ox[kk]] * wx[kk];
    kf[kk].y = kptr[oy[kk]] * wy[kk];
  }

  for (int j = 0; j < 256; ++j) {
    const int kb = j << 4;
    // S^T = K_tile(16x20) @ Q^T(20x16)
    v8f s8 = zero8();
#pragma unroll
    for (int kk = 0; kk < 5; ++kk) s8 = wmma4(kf[kk], qf[kk], s8);

    // Prefetch next tile's K fragments (clamped on last iter).
    const float* kptr_n = kptr + (size_t)((j + 1 < 256) ? 16 * ld : 0);
    v2f kfn[5];
#pragma unroll
    for (int kk = 0; kk < 5; ++kk) {
      kfn[kk].x = kptr_n[ox[kk]] * wx[kk];
      kfn[kk].y = kptr_n[oy[kk]] * wy[kk];
    }
    // Prefetch this tile's V fragments (independent of softmax).
    float vax[4], vay[4], vbx[4], vby[4];
#pragma unroll
    for (int kk = 0; kk < 4; ++kk) {
      const float* v0 = vb + (size_t)(kk * 4 + 2 * hi) * ld;
      vax[kk] = v0[r];
      vay[kk] = v0[ld + r];
      vbx[kk] = v0[dv1] * wv1;
      vby[kk] = v0[ld + dv1] * wv1;
    }
    // Key-mask: 8 contiguous floats -> two b128 loads.
    const v4f pm0 = *(const v4f*)(pm + kb + 8 * hi);
    const v4f pm1 = *(const v4f*)(pm + kb + 8 * hi + 4);

    // Online softmax (lane holds query col r, key rows v + 8*hi).
    float p[8];
    float mx = -3.0e38f;
#pragma unroll
    for (int v = 0; v < 8; ++v) {
      const float mk = (v < 4) ? pm0[v & 3] : pm1[v & 3];
      float s = s8[v] * scale;
      if (!(mk > 0.f)) s = NEGV;
      p[v] = s;
      mx = fmaxf(mx, s);
    }
    mx = fmaxf(mx, __shfl_xor(mx, 16, 32));
    const float m_new = fmaxf(m_s, mx);
    const float resc = __expf(m_s - m_new);
    float ps = 0.f;
#pragma unroll
    for (int v = 0; v < 8; ++v) {
      p[v] = __expf(p[v] - m_new);
      ps += p[v];
    }
    ps += __shfl_xor(ps, 16, 32);
    l_s = l_s * resc + ps;
    m_s = m_new;
    o0 *= resc;
    o1 *= resc;

    __syncthreads();  // single-wave WG: cheap; orders pT reuse
#pragma unroll
    for (int v = 0; v < 8; ++v) pT[(v + 8 * hi) * 16 + r] = p[v];
    __syncthreads();

    // O^T += V^T(32pad x 16keys) @ P^T(16keys x 16queries)
#pragma unroll
    for (int kk = 0; kk < 4; ++kk) {
      const int krow4 = kk * 4 + 2 * hi;
      v2f bf;
      bf.x = pT[krow4 * 16 + r];
      bf.y = pT[(krow4 + 1) * 16 + r];
      v2f a;
      a.x = vax[kk];
      a.y = vay[kk];
      o0 = wmma4(a, bf, o0);
      a.x = vbx[kk];
      a.y = vby[kk];
      o1 = wmma4(a, bf, o1);
    }

#pragma unroll
    for (int kk = 0; kk < 5; ++kk) kf[kk] = kfn[kk];
    kptr = kptr_n;
    vb += (size_t)16 * ld;
  }

  const float inv = 1.f / l_s;
#pragma unroll
  for (int v = 0; v < 8; ++v) {
    const int d = v + 8 * hi;
    outp[(size_t)(i0 + r) * 144 + h * 18 + d] = o0[v] * inv;
    const int d1 = 16 + v + 8 * hi;
    if (d1 < 18) outp[(size_t)(i0 + r) * 144 + h * 18 + d1] = o1[v] * inv;
  }
}

// ---------------------------------------------------------------------------
// Single attention with pair bias. L=64 keys, one wave per (head, 16-q tile).
// q: [64,144]; kv: [64,288] (k then v per head); biasb: [4096, 8] = pair@sa_wb.
// ---------------------------------------------------------------------------
__global__ __launch_bounds__(32) void single_attn_kernel(
    const float* __restrict__ qbuf, const float* __restrict__ kvbuf,
    const float* __restrict__ biasb, const float* __restrict__ sm,
    float* __restrict__ outp) {
  __shared__ float pT[256];
  const int h = blockIdx.y;
  const int i0 = blockIdx.x << 4;
  const int lane = threadIdx.x;
  const int hi = lane >> 4;
  const int r = lane & 15;
  const float* Q = qbuf + h * 18;         // ld 144
  const float* V = kvbuf + 144 + h * 18;  // ld 288
  const float scale = 1.0f / sqrtf(18.0f);

  int ox[5], oy[5];
  float wx[5], wy[5];
#pragma unroll
  for (int kk = 0; kk < 5; ++kk) {
    const int d0 = kk * 4 + 2 * hi;
    ox[kk] = (d0 < 18) ? d0 : 17;
    wx[kk] = (d0 < 18) ? 1.f : 0.f;
    oy[kk] = (d0 + 1 < 18) ? d0 + 1 : 17;
    wy[kk] = (d0 + 1 < 18) ? 1.f : 0.f;
  }
  const int dv1 = (16 + r < 18) ? 16 + r : 17;
  const float wv1 = (16 + r < 18) ? 1.f : 0.f;

  v2f qf[5];
  {
    const float* qrow = Q + (size_t)(i0 + r) * 144;
#pragma unroll
    for (int kk = 0; kk < 5; ++kk) {
      qf[kk].x = qrow[ox[kk]] * wx[kk];
      qf[kk].y = qrow[oy[kk]] * wy[kk];
    }
  }

  float m_s = -1e30f, l_s = 0.f;
  v8f o0 = zero8();
  v8f o1 = zero8();

  const float* kptr = kvbuf + h * 18 + (size_t)r * 288;
  const float* vb = V;
  v2f kf[5];
#pragma unroll
  for (int kk = 0; kk < 5; ++kk) {
    kf[kk].x = kptr[ox[kk]] * wx[kk];
    kf[kk].y = kptr[oy[kk]] * wy[kk];
  }

  for (int j = 0; j < 4; ++j) {
    const int kb = j << 4;
    v8f s8 = zero8();
#pragma unroll
    for (int kk = 0; kk < 5; ++kk) s8 = wmma4(kf[kk], qf[kk], s8);

    const float* kptr_n = kptr + (size_t)((j + 1 < 4) ? 16 * 288 : 0);
    v2f kfn[5];
#pragma unroll
    for (int kk = 0; kk < 5; ++kk) {
      kfn[kk].x = kptr_n[ox[kk]] * wx[kk];
      kfn[kk].y = kptr_n[oy[kk]] * wy[kk];
    }
    float vax[4], vay[4], vbx[4], vby[4];
#pragma unroll
    for (int kk = 0; kk < 4; ++kk) {
      const float* v0 = vb + (size_t)(kk * 4 + 2 * hi) * 288;
      vax[kk] = v0[r];
      vay[kk] = v0[288 + r];
      vbx[kk] = v0[dv1] * wv1;
      vby[kk] = v0[288 + dv1] * wv1;
    }
    const v4f sm0 = *(const v4f*)(sm + kb + 8 * hi);
    const v4f sm1 = *(const v4f*)(sm + kb + 8 * hi + 4);

    float p[8];
    float mx = -3.0e38f;
    const int qi = i0 + r;
#pragma unroll
    for (int v = 0; v < 8; ++v) {
      const int kj = kb + v + 8 * hi;
      const float mk = (v < 4) ? sm0[v & 3] : sm1[v & 3];
      float s = s8[v] * scale + biasb[(size_t)(qi * 64 + kj) * 8 + h];
      if (!(mk > 0.f)) s = NEGV;
      p[v] = s;
      mx = fmaxf(mx, s);
    }
    mx = fmaxf(mx, __shfl_xor(mx, 16, 32));
    const float m_new = fmaxf(m_s, mx);
    const float resc = __expf(m_s - m_new);
    float ps = 0.f;
#pragma unroll
    for (int v = 0; v < 8; ++v) {
      p[v] = __expf(p[v] - m_new);
      ps += p[v];
    }
    ps += __shfl_xor(ps, 16, 32);
    l_s = l_s * resc + ps;
    m_s = m_new;
    o0 *= resc;
    o1 *= resc;

    __syncthreads();
#pragma unroll
    for (int v = 0; v < 8; ++v) pT[(v + 8 * hi) * 16 + r] = p[v];
    __syncthreads();

#pragma unroll
    for (int kk = 0; kk < 4; ++kk) {
      const int krow4 = kk * 4 + 2 * hi;
      v2f bf;
      bf.x = pT[krow4 * 16 + r];
      bf.y = pT[(krow4 + 1) * 16 + r];
      v2f a;
      a.x = vax[kk];
      a.y = vay[kk];
      o0 = wmma4(a, bf, o0);
      a.x = vbx[kk];
      a.y = vby[kk];
      o1 = wmma4(a, bf, o1);
    }

#pragma unroll
    for (int kk = 0; kk < 5; ++kk) kf[kk] = kfn[kk];
    kptr = kptr_n;
    vb += (size_t)16 * 288;
  }

  const float inv = 1.f / l_s;
#pragma unroll
  for (int v = 0; v < 8; ++v) {
    const int d = v + 8 * hi;
    outp[(size_t)(i0 + r) * 144 + h * 18 + d] = o0[v] * inv;
    const int d1 = 16 + v + 8 * hi;
    if (d1 < 18) outp[(size_t)(i0 + r) * 144 + h * 18 + d1] = o1[v] * inv;
  }
}

// ---------------------------------------------------------------------------
// Elementwise fusions
// ---------------------------------------------------------------------------
__global__ void pair_res1_kernel(const float* __restrict__ pin,
                                 const float* __restrict__ proj,
                                 const float* __restrict__ gate,
                                 const float* __restrict__ pm,
                                 float* __restrict__ pout, int total) {
  const int i = blockIdx.x * blockDim.x + threadIdx.x;
  if (i < total) {
    const int n = i / 144;
    const float g = 1.f / (1.f + __expf(-gate[i]));
    pout[i] = pin[i] + proj[i] * g * pm[n];
  }
}

__global__ void res_mask_kernel(const float* __restrict__ base,
                                const float* __restrict__ t,
                                const float* __restrict__ mask,
                                float* __restrict__ out, int total, int D) {
  const int i = blockIdx.x * blockDim.x + threadIdx.x;
  if (i < total) out[i] = base[i] + t[i] * mask[i / D];
}

__global__ void add_mask_kernel(const float* __restrict__ t,
                                const float* __restrict__ mask,
                                float* __restrict__ out, int total, int D) {
  const int i = blockIdx.x * blockDim.x + threadIdx.x;
  if (i < total) out[i] = out[i] + t[i] * mask[i / D];
}

// ---------------------------------------------------------------------------
extern "C" void kernel_launch(void* const* d_in, const int* in_sizes, int n_in,
                              void* d_out, int out_size, void* d_ws,
                              size_t ws_size, hipStream_t stream) {
  const float* single_in = (const float*)d_in[0];   // 64*144
  const float* pair_in   = (const float*)d_in[1];   // 4096*144
  const float* sm        = (const float*)d_in[2];   // 64
  const float* pm        = (const float*)d_in[3];   // 4096
  const float* pa_ln_g   = (const float*)d_in[4];
  const float* pa_ln_b   = (const float*)d_in[5];
  const float* pa_wqkv   = (const float*)d_in[6];   // 144x432
  const float* pa_wg     = (const float*)d_in[7];   // 144x144
  const float* pa_wout   = (const float*)d_in[8];   // 144x144
  const float* pt_ln_g   = (const float*)d_in[9];
  const float* pt_ln_b   = (const float*)d_in[10];
  const float* pt_w1     = (const float*)d_in[11];  // 144x576
  const float* pt_b1     = (const float*)d_in[12];
  const float* pt_w2     = (const float*)d_in[13];  // 576x144
  const float* pt_b2     = (const float*)d_in[14];
  const float* sa_ln_g   = (const float*)d_in[15];
  const float* sa_ln_b   = (const float*)d_in[16];
  const float* sa_wq     = (const float*)d_in[17];  // 144x144
  const float* sa_wkv    = (const float*)d_in[18];  // 144x288
  const float* sa_wb     = (const float*)d_in[19];  // 144x8
  const float* sa_wout   = (const float*)d_in[20];  // 144x144
  const float* st_ln_g   = (const float*)d_in[21];
  const float* st_ln_b   = (const float*)d_in[22];
  const float* st_w1     = (const float*)d_in[23];  // 144x576
  const float* st_b1     = (const float*)d_in[24];
  const float* st_w2     = (const float*)d_in[25];  // 576x144
  const float* st_b2     = (const float*)d_in[26];

  float* out = (float*)d_out;
  float* single_out = out;       // 9216
  float* pair_out = out + 9216;  // 589824

  float* ws = (float*)d_ws;
  float* zbuf = ws;              // 4096x144 (z; later pair-transition LN)
  float* big  = zbuf + 589824;   // 4096x576 (qkv[432]; later h[576])
  float* gate = big + 2359296;   // 4096x144
  float* aout = gate + 589824;   // 4096x144 (attn out; later t2)
  float* scr  = aout + 589824;   // 4096x144 (proj; later single-track bufs)
  // single-track carve-outs (used after proj is dead)
  float* biasb  = scr;            // 4096x8
  float* sbuf   = scr + 32768;    // 64x144
  float* qbuf   = sbuf + 9216;    // 64x144
  float* kvbuf  = qbuf + 9216;    // 64x288
  float* satbuf = kvbuf + 18432;  // 64x144
  float* sprojb = satbuf + 9216;  // 64x144
  float* stbuf  = sprojb + 9216;  // 64x144
  float* shbuf  = stbuf + 9216;   // 64x576
  float* st2buf = shbuf + 36864;  // 64x144

  const dim3 b32(32);
  const dim3 b256(256);

  // ---- pair attention ----
  ln_kernel<<<4096, b32, 0, stream>>>(pair_in, zbuf, pa_ln_g, pa_ln_b, 144);
  gemm_f32_kernel<<<dim3(7, 256), b32, 0, stream>>>(zbuf, 144, pa_wqkv, 432,
                                                    big, 432, 4096, 432, 144,
                                                    nullptr, 0);
  gemm_f32_kernel<<<dim3(3, 256), b32, 0, stream>>>(zbuf, 144, pa_wg, 144,
                                                    gate, 144, 4096, 144, 144,
                                                    nullptr, 0);
  pair_attn_kernel<<<dim3(256, 8), b32, 0, stream>>>(big, pm, aout);
  gemm_f32_kernel<<<dim3(3, 256), b32, 0, stream>>>(aout, 144, pa_wout, 144,
                                                    scr, 144, 4096, 144, 144,
                                                    nullptr, 0);
  pair_res1_kernel<<<(589824 + 255) / 256, b256, 0, stream>>>(
      pair_in, scr, gate, pm, pair_out, 589824);

  // ---- pair transition ----
  ln_kernel<<<4096, b32, 0, stream>>>(pair_out, zbuf, pt_ln_g, pt_ln_b, 144);
  gemm_f32_kernel<<<dim3(9, 256), b32, 0, stream>>>(zbuf, 144, pt_w1, 576,
                                                    big, 576, 4096, 576, 144,
                                                    pt_b1, 1);
  gemm_f32_kernel<<<dim3(3, 256), b32, 0, stream>>>(big, 576, pt_w2, 144,
                                                    aout, 144, 4096, 144, 576,
                                                    pt_b2, 0);
  add_mask_kernel<<<(589824 + 255) / 256, b256, 0, stream>>>(aout, pm,
                                                             pair_out, 589824,
                                                             144);

  // ---- single attention (pair bias from final pair) ----
  ln_kernel<<<64, b32, 0, stream>>>(single_in, sbuf, sa_ln_g, sa_ln_b, 144);
  gemm_f32_kernel<<<dim3(3, 4), b32, 0, stream>>>(sbuf, 144, sa_wq, 144, qbuf,
                                                  144, 64, 144, 144, nullptr,
                                                  0);
  gemm_f32_kernel<<<dim3(5, 4), b32, 0, stream>>>(sbuf, 144, sa_wkv, 288,
                                                  kvbuf, 288, 64, 288, 144,
                                                  nullptr, 0);
  gemm_f32_kernel<<<dim3(1, 256), b32, 0, stream>>>(pair_out, 144, sa_wb, 8,
                                                    biasb, 8, 4096, 8, 144,
                                                    nullptr, 0);
  single_attn_kernel<<<dim3(4, 8), b32, 0, stream>>>(qbuf, kvbuf, biasb, sm,
                                                     satbuf);
  gemm_f32_kernel<<<dim3(3, 4), b32, 0, stream>>>(satbuf, 144, sa_wout, 144,
                                                  sprojb, 144, 64, 144, 144,
                                                  nullptr, 0);
  res_mask_kernel<<<(9216 + 255) / 256, b256, 0, stream>>>(
      single_in, sprojb, sm, single_out, 9216, 144);

  // ---- single transition ----
  ln_kernel<<<64, b32, 0, stream>>>(single_out, stbuf, st_ln_g, st_ln_b, 144);
  gemm_f32_kernel<<<dim3(9, 4), b32, 0, stream>>>(stbuf, 144, st_w1, 576,
                                                  shbuf, 576, 64, 576, 144,
                                                  st_b1, 1);
  gemm_f32_kernel<<<dim3(3, 4), b32, 0, stream>>>(shbuf, 576, st_w2, 144,
                                                  st2buf, 144, 64, 144, 576,
                                                  st_b2, 0);
  add_mask_kernel<<<(9216 + 255) / 256, b256, 0, stream>>>(st2buf, sm,
                                                           single_out, 9216,
                                                           144);
}